// SAGEConvReg_6536940224566
// MI455X (gfx1250) — compile-verified
//
#include <hip/hip_runtime.h>

typedef __attribute__((ext_vector_type(2))) float v2f;
typedef __attribute__((ext_vector_type(8))) float v8f;

// ---------------------------------------------------------------------------
// Kernel 1: zero the workspace accumulators (acc, deg, colsum, loss)
// ---------------------------------------------------------------------------
__global__ void zero_ws_kernel(float* __restrict__ ws, long n) {
    long i = (long)blockIdx.x * blockDim.x + threadIdx.x;
    long stride = (long)gridDim.x * blockDim.x;
    for (; i < n; i += stride) ws[i] = 0.0f;
}

// ---------------------------------------------------------------------------
// Kernel 2: edge scatter. One thread per (edge, feature). Within a wave all 32
// lanes process the same edge with consecutive features -> 128B coalesced
// gather from x and 128B coalesced fp32 atomic burst into acc (acc fits in L2).
// ---------------------------------------------------------------------------
__global__ void scatter_edges_kernel(const float* __restrict__ x,
                                     const int* __restrict__ src_idx,
                                     const int* __restrict__ dst_idx,
                                     float* __restrict__ acc,
                                     float* __restrict__ deg,
                                     long n_edges) {
    long gid = (long)blockIdx.x * blockDim.x + threadIdx.x;
    long total = n_edges << 6;               // D == 64
    if (gid >= total) return;
    long e = gid >> 6;
    int  j = (int)(gid & 63);
    int  s = src_idx[e];
    int  d = dst_idx[e];
    atomicAdd(&acc[(long)d * 64 + j], x[(long)s * 64 + j]);
    if (j == 0) atomicAdd(&deg[d], 1.0f);
}

// ---------------------------------------------------------------------------
// Kernel 3: y_mean = acc / max(deg,1) in place, plus per-block column partial
// sums reduced through LDS, one fp32 atomic per column per block.
// blockDim = 1024 -> 16 rows per block.
// ---------------------------------------------------------------------------
__global__ void mean_colsum_kernel(float* __restrict__ acc,
                                   const float* __restrict__ deg,
                                   float* __restrict__ colsum,
                                   int n_dst) {
    __shared__ float cs[64];
    int tid = threadIdx.x;
    if (tid < 64) cs[tid] = 0.0f;
    __syncthreads();
    int row = blockIdx.x * 16 + (tid >> 6);
    int j   = tid & 63;
    if (row < n_dst) {
        float dg = deg[row];
        float ym = acc[(long)row * 64 + j] / fmaxf(dg, 1.0f);
        acc[(long)row * 64 + j] = ym;      // overwrite acc with y_mean
        atomicAdd(&cs[j], ym);             // ds_add_f32
    }
    __syncthreads();
    if (tid < 64) atomicAdd(&colsum[tid], cs[tid]);
}

// ---------------------------------------------------------------------------
// Kernel 4: h = [x_dst | y_mean] @ W_w^T + b via V_WMMA_F32_16X16X4_F32.
// One wave computes a 16-row strip across all 64 output columns:
//   4 accumulators (16x16), K = 128 in steps of 4 -> 128 WMMA ops/wave.
// Fragment layouts (ISA 7.12.2, 32-bit):
//   A 16x4 : lane L -> M = L%16, VGPR0/1 = A[M][k0 + (L/16)*2 + {0,1}]
//   B 4x16 : lane L -> N = L%16, VGPR0/1 = B[k0 + (L/16)*2 + {0,1}][N]
//   C 16x16: VGPR v, lane L -> M = v + (L/16)*8, N = L%16
// B[k][n] = W_w[n][k] (W_w is row-major 64x128), so each B fragment load is a
// contiguous float2 from W_w's row n.
// ---------------------------------------------------------------------------
__global__ void gemm_wmma_kernel(const float* __restrict__ x,     // (n_src,64)
                                 const float* __restrict__ ymean, // (n_dst,64)
                                 const float* __restrict__ Ww,    // (64,128)
                                 const float* __restrict__ Wb,    // (64)
                                 float* __restrict__ h,           // (n_dst,64)
                                 int n_dst) {
    int wave    = blockIdx.x * (blockDim.x >> 5) + (threadIdx.x >> 5);
    int nstrips = (n_dst + 15) >> 4;
    if (wave >= nstrips) return;             // uniform per wave: EXEC stays all-1s

    int lane = threadIdx.x & 31;
    int m    = lane & 15;                    // M (A) / N (B,C)
    int hsel = lane >> 4;                    // which K-pair half
    int row  = wave * 16 + m;
    int arow = row < n_dst ? row : n_dst - 1; // clamped load row (EXEC uniform)

    const float* xrow = x     + (long)arow * 64;
    const float* yrow = ymean + (long)arow * 64;
    const float* w0 = Ww + (long)(m)      * 128;
    const float* w1 = Ww + (long)(16 + m) * 128;
    const float* w2 = Ww + (long)(32 + m) * 128;
    const float* w3 = Ww + (long)(48 + m) * 128;

    v8f c0 = {}, c1 = {}, c2 = {}, c3 = {};
    int kb = hsel * 2;

#pragma unroll
    for (int k0 = 0; k0 < 128; k0 += 4) {
        int k = k0 + kb;
        v2f a = (k0 < 64) ? *(const v2f*)(xrow + k)
                          : *(const v2f*)(yrow + (k - 64));
        v2f b0 = *(const v2f*)(w0 + k);
        v2f b1 = *(const v2f*)(w1 + k);
        v2f b2 = *(const v2f*)(w2 + k);
        v2f b3 = *(const v2f*)(w3 + k);
        c0 = __builtin_amdgcn_wmma_f32_16x16x4_f32(false, a, false, b0, (short)0, c0, false, false);
        c1 = __builtin_amdgcn_wmma_f32_16x16x4_f32(false, a, false, b1, (short)0, c1, false, false);
        c2 = __builtin_amdgcn_wmma_f32_16x16x4_f32(false, a, false, b2, (short)0, c2, false, false);
        c3 = __builtin_amdgcn_wmma_f32_16x16x4_f32(false, a, false, b3, (short)0, c3, false, false);
    }

    float bias0 = Wb[m];
    float bias1 = Wb[16 + m];
    float bias2 = Wb[32 + m];
    float bias3 = Wb[48 + m];

    int mbase = wave * 16 + hsel * 8;
#pragma unroll
    for (int v = 0; v < 8; ++v) {
        int M = mbase + v;
        if (M < n_dst) {
            long o = (long)M * 64 + m;
            h[o]      = c0[v] + bias0;
            h[o + 16] = c1[v] + bias1;
            h[o + 32] = c2[v] + bias2;
            h[o + 48] = c3[v] + bias3;
        }
    }
}

// ---------------------------------------------------------------------------
// Kernel 5: reg-loss partial sums. Grid-stride over n_src*64, wave shuffle
// reduce -> LDS -> one global atomic per block.
// ---------------------------------------------------------------------------
__global__ void reg_loss_kernel(const float* __restrict__ x,
                                const float* __restrict__ u_sum,
                                const float* __restrict__ colsum,
                                float* __restrict__ loss,
                                long n_src, float inv_ndst) {
    __shared__ float sacc;
    if (threadIdx.x == 0) sacc = 0.0f;
    __syncthreads();

    long total  = n_src << 6;                // D == 64
    long i      = (long)blockIdx.x * blockDim.x + threadIdx.x;
    long stride = (long)gridDim.x * blockDim.x;
    float local = 0.0f;
    for (; i < total; i += stride) {
        long r = i >> 6;
        int  j = (int)(i & 63);
        float mx = colsum[j] * inv_ndst;     // mean_x[j]
        float mu = u_sum[r] * inv_ndst;      // mean_u[r]
        float d  = mu * x[i] - mx;
        local += d * d;
    }
    // wave32 shuffle reduction
    for (int off = 16; off > 0; off >>= 1)
        local += __shfl_down(local, off, 32);
    if ((threadIdx.x & 31) == 0) atomicAdd(&sacc, local);
    __syncthreads();
    if (threadIdx.x == 0) atomicAdd(loss, sacc);
}

__global__ void finalize_loss_kernel(const float* __restrict__ loss,
                                     float* __restrict__ out, float scale) {
    if (blockIdx.x == 0 && threadIdx.x == 0) out[0] = loss[0] * scale;
}

// ---------------------------------------------------------------------------
// Launch
// ---------------------------------------------------------------------------
extern "C" void kernel_launch(void* const* d_in, const int* in_sizes, int n_in,
                              void* d_out, int out_size, void* d_ws, size_t ws_size,
                              hipStream_t stream) {
    const float* x       = (const float*)d_in[0];
    // d_in[1] = w : only feeds y_wsum, which is dead in the reference -> skip
    const float* u_sum   = (const float*)d_in[2];
    const float* Ww      = (const float*)d_in[3];
    const float* Wb      = (const float*)d_in[4];
    const int*   src_idx = (const int*)d_in[5];
    const int*   dst_idx = (const int*)d_in[6];
    // d_in[7] holds n_dst on device; derive it on host (graph-capture safe):
    const int  n_classes = in_sizes[4];                  // 64
    const int  n_dst     = (out_size - 1) / n_classes;   // 50000
    const long n_src     = in_sizes[2];                  // 100000
    const long n_edges   = in_sizes[1];                  // 1,000,000

    // workspace layout (floats): acc | deg | colsum(64) | loss(1)
    float* ws     = (float*)d_ws;
    float* acc    = ws;
    float* deg    = acc + (long)n_dst * 64;
    float* colsum = deg + n_dst;
    float* loss   = colsum + 64;
    long   zn     = (long)n_dst * 64 + n_dst + 65;

    float* h        = (float*)d_out;
    float* out_loss = h + (long)n_dst * (long)n_classes;

    zero_ws_kernel<<<2048, 256, 0, stream>>>(ws, zn);

    long scat_threads = n_edges << 6;
    int  scat_blocks  = (int)((scat_threads + 255) / 256);
    scatter_edges_kernel<<<scat_blocks, 256, 0, stream>>>(
        x, src_idx, dst_idx, acc, deg, n_edges);

    int mean_blocks = (n_dst + 15) / 16;
    mean_colsum_kernel<<<mean_blocks, 1024, 0, stream>>>(acc, deg, colsum, n_dst);

    int nstrips     = (n_dst + 15) / 16;          // 3125
    int gemm_blocks = (nstrips + 7) / 8;          // 8 waves (256 threads) per block
    gemm_wmma_kernel<<<gemm_blocks, 256, 0, stream>>>(x, acc, Ww, Wb, h, n_dst);

    float inv_ndst = 1.0f / (float)n_dst;
    reg_loss_kernel<<<2048, 256, 0, stream>>>(x, u_sum, colsum, loss, n_src, inv_ndst);

    finalize_loss_kernel<<<1, 32, 0, stream>>>(
        loss, out_loss, 1.0f / ((float)n_src * (float)n_classes));
}